// MultiCodeVectorQuantizer_5257039970377
// MI455X (gfx1250) — compile-verified
//
#include <hip/hip_runtime.h>

typedef __attribute__((ext_vector_type(2))) float v2f;
typedef __attribute__((ext_vector_type(8))) float v8f;
typedef __attribute__((address_space(3))) float lds_f;

#define C        256        // embedding dim
#define C4       64         // float4s per row
#define NCODES   8192       // codebook rows
#define QROWS    16384      // 16*1024 query rows
#define RPB      64         // query rows per block (4 waves x 16)
#define TILE     16         // codebook rows per WMMA tile
#define NTILES   (NCODES / TILE)   // 512
#define LSTRIDE  260        // padded LDS row stride (floats): 4-bank skew per row
#define PART_OFF NCODES     // ws float offset of block partials (esq occupies [0,8192))

// ---- precompute ||e_n||^2 for all codebook rows (one-off, 8.4 MB read) ----
__global__ __launch_bounds__(256) void esq_kernel(const float* __restrict__ e,
                                                  float* __restrict__ esq)
{
    int wave = threadIdx.x >> 5, lane = threadIdx.x & 31;
    int row  = blockIdx.x * 8 + wave;            // 1024 blocks x 8 waves = 8192 rows
    const float4* er = (const float4*)(e + (size_t)row * C);
    float s = 0.f;
    #pragma unroll
    for (int i = 0; i < 2; ++i) {
        float4 v = er[lane + 32 * i];            // coalesced: 64 float4 per row
        s += v.x*v.x + v.y*v.y + v.z*v.z + v.w*v.w;
    }
    #pragma unroll
    for (int off = 16; off >= 1; off >>= 1)
        s += __shfl_xor(s, off, 32);
    if (lane == 0) esq[row] = s;
}

__global__ __launch_bounds__(128) void vq_main(const float* __restrict__ q,
                                               const float* __restrict__ e,
                                               const float* __restrict__ esq_g,
                                               float* __restrict__ out,
                                               float* __restrict__ partial)
{
    __shared__ float qs[RPB * LSTRIDE];          // 64 x 260 f32 = 66560 B
    __shared__ float es[2][TILE * LSTRIDE];      // double-buffered E tile, 2x16640 B
    __shared__ int   bestidx[RPB];
    __shared__ float lsum[128];

    const int tid  = threadIdx.x;
    const int wave = tid >> 5;
    const int lane = tid & 31;
    const int l16  = lane & 15;
    const int hi   = lane >> 4;                  // half-wave: K pair 0,1 vs 2,3
    const int row0 = blockIdx.x * RPB;

    // LDS byte offset of es[0][0] for async-to-LDS destinations
    const unsigned es_lds = (unsigned)(uintptr_t)(lds_f*)&es[0][0];

    // issue one tile's 16x256 f32 as 1024 async b128 global->LDS copies (8/thread)
    auto issue_tile = [&](int t, int buf) {
        const float* gsrc = e + (size_t)t * TILE * C;
        const unsigned lbase = es_lds + (unsigned)buf * (TILE * LSTRIDE * 4);
        if (t + 1 < NTILES)  // warm L2 for the tile after next
            __builtin_prefetch(e + (size_t)(t + 1) * TILE * C + tid * 32, 0, 3);
        for (int i = tid; i < TILE * C4; i += 128) {
            int r = i >> 6, c4 = i & 63;
            unsigned loff = lbase + (unsigned)(r * LSTRIDE + c4 * 4) * 4u;
            unsigned long long ga = (unsigned long long)(uintptr_t)(gsrc + i * 4);
            asm volatile("global_load_async_to_lds_b128 %0, %1, off"
                         :: "v"(loff), "v"(ga) : "memory");
        }
    };

    // ---- stage this block's 64 query rows into LDS + kick off tile 0 ----
    for (int i = tid; i < RPB * C4; i += 128) {
        int r = i >> 6, c4 = i & 63;
        float4 v = ((const float4*)q)[(size_t)(row0 + r) * C4 + c4];
        *(float4*)&qs[r * LSTRIDE + c4 * 4] = v;
    }
    issue_tile(0, 0);
    asm volatile("s_wait_asynccnt 0x0" ::: "memory");
    __syncthreads();

    // A fragment source: query row M = wave*16 + (lane&15), K pair per half-wave
    const float* aptr = &qs[(wave * TILE + l16) * LSTRIDE + hi * 2];

    // running (min dist', argmin) per C-slot: slot v <-> row M = v + 8*hi,
    // lane%16 <-> n%16; tiles scanned in order -> first-occurrence minima.
    float mind[8]; int mini[8];
    #pragma unroll
    for (int v = 0; v < 8; ++v) { mind[v] = 3.4e38f; mini[v] = 0; }

    for (int t = 0; t < NTILES; ++t) {
        const int buf = t & 1;
        if (t + 1 < NTILES) issue_tile(t + 1, buf ^ 1);   // overlap next stage
        float esq_n = esq_g[t * TILE + l16];              // L2-hot dword

        // 16x16 tile of Q.E^T via 64 chained fp32 WMMAs (K = 256)
        const float* bp = &es[buf][l16 * LSTRIDE + hi * 2];
        v8f acc = {};
        #pragma unroll
        for (int k = 0; k < C; k += 4) {
            v2f a = *(const v2f*)(aptr + k);   // ds_load_2addr_b64, conflict-free
            v2f b = *(const v2f*)(bp + k);
            acc = __builtin_amdgcn_wmma_f32_16x16x4_f32(false, a, false, b,
                                                        (short)0, acc, false, false);
        }

        // dist' = ||e_n||^2 - 2 q.e  (||q||^2 constant per row -> drop for argmin)
        int nidx = t * TILE + l16;
        #pragma unroll
        for (int v = 0; v < 8; ++v) {
            float d = __builtin_fmaf(-2.0f, acc[v], esq_n);
            if (d < mind[v]) { mind[v] = d; mini[v] = nidx; }
        }

        asm volatile("s_wait_asynccnt 0x0" ::: "memory");  // next buffer landed
        __syncthreads();                                   // ... in every wave
    }

    // cross-lane argmin within each 16-lane half (xor butterfly, width 16),
    // ties -> smaller index (jnp.argmin first-occurrence semantics)
    #pragma unroll
    for (int v = 0; v < 8; ++v) {
        float d = mind[v]; int ix = mini[v];
        #pragma unroll
        for (int off = 8; off >= 1; off >>= 1) {
            float od = __shfl_xor(d, off, 16);
            int   oi = __shfl_xor(ix, off, 16);
            if (od < d || (od == d && oi < ix)) { d = od; ix = oi; }
        }
        if (l16 == 0) {
            int rloc = wave * TILE + hi * 8 + v;   // block-local query row
            bestidx[rloc] = ix;
            ((int*)out)[row0 + rloc] = ix;         // output 0: code_indices (int32)
        }
    }
    __syncthreads();

    // gather quantized rows (== straight-through output) + deterministic loss partial
    float accl = 0.0f;
    float* outq = out + QROWS;
    for (int i = tid; i < RPB * C4; i += 128) {
        int r = i >> 6, c4 = i & 63;
        float4 ev = ((const float4*)e)[(size_t)bestidx[r] * C4 + c4];
        float4 qv = *(const float4*)&qs[r * LSTRIDE + c4 * 4];
        ((float4*)outq)[(size_t)(row0 + r) * C4 + c4] = ev;
        float dx = qv.x - ev.x, dy = qv.y - ev.y;
        float dz = qv.z - ev.z, dw = qv.w - ev.w;
        accl += dx*dx + dy*dy + dz*dz + dw*dw;
    }
    lsum[tid] = accl;
    __syncthreads();
    for (int s = 64; s > 0; s >>= 1) {
        if (tid < s) lsum[tid] += lsum[tid + s];
        __syncthreads();
    }
    if (tid == 0) partial[blockIdx.x] = lsum[0];
}

// fixed-order final reduction: replay-deterministic (no float atomics)
__global__ __launch_bounds__(256) void vq_loss_finalize(const float* __restrict__ partial,
                                                        float* __restrict__ out)
{
    __shared__ float s[256];
    int tid = threadIdx.x;
    s[tid] = partial[tid];          // exactly 256 block partials
    __syncthreads();
    for (int k = 128; k > 0; k >>= 1) {
        if (tid < k) s[tid] += s[tid + k];
        __syncthreads();
    }
    if (tid == 0) {
        // vq_loss = codebook + 0.25*commit = 1.25 * mean((q - e[idx])^2)
        out[(size_t)QROWS + (size_t)QROWS * C] =
            1.25f * s[0] / ((float)QROWS * (float)C);
    }
}

extern "C" void kernel_launch(void* const* d_in, const int* in_sizes, int n_in,
                              void* d_out, int out_size, void* d_ws, size_t ws_size,
                              hipStream_t stream)
{
    const float* q = (const float*)d_in[0];   // [16,1024,256] f32
    const float* e = (const float*)d_in[1];   // [8192,256]   f32
    float* out     = (float*)d_out;           // idx[16384] | quant[16384*256] | loss[1]
    float* esq     = (float*)d_ws;            // 8192 floats
    float* partial = esq + PART_OFF;          // 256 floats

    esq_kernel<<<NCODES / 8, 256, 0, stream>>>(e, esq);
    vq_main<<<QROWS / RPB, 128, 0, stream>>>(q, e, esq, out, partial);
    vq_loss_finalize<<<1, 256, 0, stream>>>(partial, out);
}